// GCNAggregator_386547056910
// MI455X (gfx1250) — compile-verified
//
#include <hip/hip_runtime.h>
#include <hip/hip_bf16.h>

// Problem constants (fixed by the reference)
#define BB 4096
#define KK 32
#define UU 16384
#define DD 256

typedef float v2f __attribute__((ext_vector_type(2)));
typedef float v8f __attribute__((ext_vector_type(8)));

// ---------------------------------------------------------------------------
// Kernel 0: zero the column-degree histogram in workspace
// ---------------------------------------------------------------------------
__global__ void gcn_zero_kernel(int* __restrict__ p, int n) {
    int i = blockIdx.x * blockDim.x + threadIdx.x;
    if (i < n) p[i] = 0;
}

// ---------------------------------------------------------------------------
// Kernel 1: one wave32 per row. Per-row dedupe (set semantics) via uniform
// shuffle loop, ballot -> first-occurrence bitmask, atomic column counts.
// ---------------------------------------------------------------------------
__global__ void gcn_count_kernel(const int* __restrict__ neigh,
                                 int* __restrict__ colcnt,
                                 unsigned int* __restrict__ firstmask) {
    int gtid = blockIdx.x * blockDim.x + threadIdx.x;
    int row  = gtid >> 5;          // wave32: one row per wave
    int lane = gtid & 31;
    int v = neigh[row * KK + lane];

    bool first = true;
    // Uniform loop: every lane executes all 32 iterations (shfl sources active)
    for (int t = 0; t < 32; ++t) {
        int ov = __shfl(v, t, 32);
        if (t < lane && ov == v) first = false;
    }
    unsigned long long bal = __ballot(first);   // wave32: low 32 bits
    unsigned int fm = (unsigned int)bal;
    if (lane == 0) firstmask[row] = fm;
    if (first) atomicAdd(&colcnt[v], 1);
}

// ---------------------------------------------------------------------------
// Kernel 2: WMMA gather-accumulate.
// Block = 16 output rows (M-tile), 256 threads = 8 waves.
// K' = concatenated 16x32 = 512 neighbor slots; A is block-diagonal weights
// (duplicates weighted 0 => dedupe handled), B gathered from the tables.
// Each wave owns tasks (table, n-tile): 2 tables x 16 n-tiles = 32 tasks.
// Per task: 128 x V_WMMA_F32_16X16X4_F32 accumulating into v8f.
// ---------------------------------------------------------------------------
__global__ void __launch_bounds__(256)
gcn_wmma_kernel(const int* __restrict__ neigh,
                const float* __restrict__ feats,
                const float* __restrict__ noise,
                const int* __restrict__ colcnt,
                const unsigned int* __restrict__ firstmask,
                float* __restrict__ out) {
    __shared__ int   lds_idx[16 * KK];   // 2 KB
    __shared__ float lds_w[16 * KK];     // 2 KB

    const int m0  = blockIdx.x * 16;
    const int tid = threadIdx.x;

    // Stage neighbor ids + normalized weights for this 16-row tile.
    // weight(m,k) = first_occurrence ? rsqrt(rowdeg) * rsqrt(coldeg[j]) : 0
    for (int e = tid; e < 16 * KK; e += 256) {
        int m = e >> 5, k = e & 31;
        int gb = m0 + m;
        int j  = neigh[gb * KK + k];           // coalesced
        unsigned int fm = firstmask[gb];
        float w = 0.0f;
        if ((fm >> k) & 1u) {
            float rd = (float)__popc(fm);      // row degree (unique count, >=1)
            float cd = (float)colcnt[j];       // col degree (>=1: row itself)
            w = rsqrtf(rd) * rsqrtf(cd);
        }
        lds_idx[e] = j;
        lds_w[e]   = w;
    }
    __syncthreads();

    const int wave = tid >> 5;
    const int lane = tid & 31;
    const int half = lane >> 4;    // 0: lanes 0-15, 1: lanes 16-31
    const int ln   = lane & 15;    // N / M position within half

    for (int t = wave; t < 32; t += 8) {
        const int table = t >> 4;
        const int n0    = (t & 15) << 4;
        const float* __restrict__ src = table ? noise : feats;
        float* __restrict__ dst = out + table * (BB * DD);

        v8f acc = {0.f, 0.f, 0.f, 0.f, 0.f, 0.f, 0.f, 0.f};

        // K' loop: 16 source rows x 8 chunks of 4
        #pragma unroll 4
        for (int chunk = 0; chunk < 128; ++chunk) {
            const int ms   = chunk >> 3;                       // which M row owns this chunk
            const int base = ms * KK + ((chunk & 7) << 2) + half * 2;

            // Broadcast (uniform per half-wave) index + weight pairs from LDS
            int   j0 = lds_idx[base],  j1 = lds_idx[base + 1];
            float w0 = lds_w[base],    w1 = lds_w[base + 1];

            // A-matrix 16x4 f32 layout: VGPR0 = K0(l<16)/K2(l>=16), VGPR1 = K1/K3.
            // Only lane ln==ms carries the (nonzero) weight -> block-diagonal A.
            const bool on = (ln == ms);
            v2f a;
            a.x = on ? w0 : 0.0f;
            a.y = on ? w1 : 0.0f;

            // B-matrix 4x16 f32: row K striped across lanes; 64B contiguous per half.
            v2f b;
            b.x = src[j0 * DD + n0 + ln];
            b.y = src[j1 * DD + n0 + ln];

            acc = __builtin_amdgcn_wmma_f32_16x16x4_f32(
                /*neg_a=*/false, a, /*neg_b=*/false, b,
                /*c_mod=*/(short)0, acc, /*reuse_a=*/false, /*reuse_b=*/false);
        }

        // C/D 16x16 f32 layout: VGPR r -> M=r (lanes 0-15) / M=r+8 (lanes 16-31)
        #pragma unroll
        for (int r = 0; r < 8; ++r) {
            int row = m0 + r + half * 8;
            dst[row * DD + n0 + ln] = acc[r];
        }
    }
}

// ---------------------------------------------------------------------------
extern "C" void kernel_launch(void* const* d_in, const int* in_sizes, int n_in,
                              void* d_out, int out_size, void* d_ws, size_t ws_size,
                              hipStream_t stream) {
    const int*   neigh = (const int*)d_in[0];     // [B, K] int32
    const float* feats = (const float*)d_in[1];   // [U, D] f32
    const float* noise = (const float*)d_in[2];   // [U, D] f32
    float*       out   = (float*)d_out;           // [B, D] + [B, D] concatenated

    int*          colcnt    = (int*)d_ws;                               // U ints (64 KB)
    unsigned int* firstmask = (unsigned int*)((char*)d_ws + UU * 4);    // B uints (16 KB)

    // 1) zero column-degree histogram
    gcn_zero_kernel<<<(UU + 255) / 256, 256, 0, stream>>>(colcnt, UU);

    // 2) per-row dedupe + column degrees (one wave32 per row)
    gcn_count_kernel<<<(BB * 32) / 256, 256, 0, stream>>>(neigh, colcnt, firstmask);

    // 3) WMMA gather-accumulate: 16 output rows per block
    gcn_wmma_kernel<<<BB / 16, 256, 0, stream>>>(neigh, feats, noise,
                                                 colcnt, firstmask, out);
}